// DVHLoss_11347303596534
// MI455X (gfx1250) — compile-verified
//
#include <hip/hip_runtime.h>

typedef float v2f __attribute__((ext_vector_type(2)));
typedef float v8f __attribute__((ext_vector_type(8)));

#define NCH   10
#define CHPAD 16
// ws float layout: [0,512) P nums (32 bins x 16 chpad), [512,1024) T nums, [1024,1056) nv (2 x 16)

__global__ void dvh_init(float* ws) {
    int t = blockIdx.x * blockDim.x + threadIdx.x;
    if (t < 1056) ws[t] = 0.0f;
}

__device__ __forceinline__ float sigm(float s, float t) {
    // sigmoid((d - s/32)*32) with t = 32*d : 1/(1+exp(s-t))
    return __builtin_amdgcn_rcpf(1.0f + __expf(s - t));
}

__global__ __launch_bounds__(256)
void dvh_main(const float* __restrict__ pred, const float* __restrict__ tgt,
              const float* __restrict__ mask, float* __restrict__ ws, int V) {
    const int lane   = threadIdx.x & 31;
    const int wave   = blockIdx.x * (blockDim.x >> 5) + (threadIdx.x >> 5);
    const int nwaves = gridDim.x * (blockDim.x >> 5);
    const int n      = blockIdx.y;
    const int ch     = lane & 15;                  // channel (B column / acc column)
    const bool hi    = (lane >= 16);
    const float s0   = (float)ch;                  // bins 0..15 tile
    const float s1   = s0 + 16.0f;                 // bins 16..31 tile

    // per-batch base pointers -> SGPR base + 32-bit VGPR offsets in the loop
    const float* __restrict__ predB = pred + (long)n * V;
    const float* __restrict__ tgtB  = tgt  + (long)n * V;
    const float* __restrict__ maskB = mask + (long)n * V * NCH;

    // loop-invariant channel clamp + select (no per-iteration EXEC branching)
    const int   chm   = (ch < NCH) ? ch : 0;
    const float chsel = (ch < NCH) ? 1.0f : 0.0f;

    v8f accP0 = {}; v8f accP1 = {}; v8f accT0 = {}; v8f accT1 = {};
    float nvacc = 0.0f;

    const int nchunks = V >> 3;                    // 8 voxels (2 x K=4) per iteration
    for (int chunk = wave; chunk < nchunks; chunk += nwaves) {
        const int k0 = (chunk << 3) + (hi ? 2 : 0);   // quad A: lanes 0-15 K0/K1, 16-31 K2/K3
        const int k1 = k0 + 4;                        // quad B

        const float2 dpA = *(const float2*)(predB + k0);
        const float2 dpB = *(const float2*)(predB + k1);
        const float2 dtA = *(const float2*)(tgtB  + k0);
        const float2 dtB = *(const float2*)(tgtB  + k1);

        const int mb0 = k0 * NCH + chm;
        const int mb1 = k1 * NCH + chm;
        v2f bA, bB;
        bA.x = maskB[mb0]       * chsel;
        bA.y = maskB[mb0 + NCH] * chsel;
        bB.x = maskB[mb1]       * chsel;
        bB.y = maskB[mb1 + NCH] * chsel;
        nvacc += (bA.x + bA.y) + (bB.x + bB.y);

        const float tpA0 = dpA.x * 32.0f, tpA1 = dpA.y * 32.0f;
        const float tpB0 = dpB.x * 32.0f, tpB1 = dpB.y * 32.0f;
        const float ttA0 = dtA.x * 32.0f, ttA1 = dtA.y * 32.0f;
        const float ttB0 = dtB.x * 32.0f, ttB1 = dtB.y * 32.0f;

        v2f a;
        // quad A
        a.x = sigm(s0, tpA0); a.y = sigm(s0, tpA1);
        accP0 = __builtin_amdgcn_wmma_f32_16x16x4_f32(false, a, false, bA, (short)0, accP0, false, false);
        a.x = sigm(s1, tpA0); a.y = sigm(s1, tpA1);
        accP1 = __builtin_amdgcn_wmma_f32_16x16x4_f32(false, a, false, bA, (short)0, accP1, false, false);
        a.x = sigm(s0, ttA0); a.y = sigm(s0, ttA1);
        accT0 = __builtin_amdgcn_wmma_f32_16x16x4_f32(false, a, false, bA, (short)0, accT0, false, false);
        a.x = sigm(s1, ttA0); a.y = sigm(s1, ttA1);
        accT1 = __builtin_amdgcn_wmma_f32_16x16x4_f32(false, a, false, bA, (short)0, accT1, false, false);
        // quad B
        a.x = sigm(s0, tpB0); a.y = sigm(s0, tpB1);
        accP0 = __builtin_amdgcn_wmma_f32_16x16x4_f32(false, a, false, bB, (short)0, accP0, false, false);
        a.x = sigm(s1, tpB0); a.y = sigm(s1, tpB1);
        accP1 = __builtin_amdgcn_wmma_f32_16x16x4_f32(false, a, false, bB, (short)0, accP1, false, false);
        a.x = sigm(s0, ttB0); a.y = sigm(s0, ttB1);
        accT0 = __builtin_amdgcn_wmma_f32_16x16x4_f32(false, a, false, bB, (short)0, accT0, false, false);
        a.x = sigm(s1, ttB0); a.y = sigm(s1, ttB1);
        accT1 = __builtin_amdgcn_wmma_f32_16x16x4_f32(false, a, false, bB, (short)0, accT1, false, false);
    }

    // writeout: C/D layout -> VGPR r, lanes 0-15: (M=r, N=lane); lanes 16-31: (M=r+8, N=lane-16)
    float* wsP  = ws;
    float* wsT  = ws + 512;
    float* wsNV = ws + 1024;
    const int rowoff = hi ? 8 : 0;
#pragma unroll
    for (int r = 0; r < 8; ++r) {
        const int bin = r + rowoff;
        atomicAdd(&wsP[bin * CHPAD + ch],        accP0[r]);
        atomicAdd(&wsP[(bin + 16) * CHPAD + ch], accP1[r]);
        atomicAdd(&wsT[bin * CHPAD + ch],        accT0[r]);
        atomicAdd(&wsT[(bin + 16) * CHPAD + ch], accT1[r]);
    }
    if (ch < NCH) atomicAdd(&wsNV[n * CHPAD + ch], nvacc);
}

__global__ void dvh_final(const float* __restrict__ ws, float* __restrict__ out) {
    __shared__ float red[512];
    const int t = threadIdx.x;
    float v = 0.0f;
    if (t < 320) {
        const int b = t / NCH, c = t % NCH;
        const float P   = ws[b * CHPAD + c];
        const float T   = ws[512 + b * CHPAD + c];
        const float nv0 = ws[1024 + c] + 1.0f;
        const float nv1 = ws[1024 + CHPAD + c] + 1.0f;
        const float d   = P - T;
        v = d * d * (1.0f / (nv0 * nv0) + 1.0f / (nv1 * nv1));
    }
    red[t] = v;
    __syncthreads();
    for (int s = 256; s > 0; s >>= 1) {
        if (t < s) red[t] += red[t + s];
        __syncthreads();
    }
    if (t == 0) out[0] = red[0] * (1.0f / 1280.0f);   // /(N*B*C) then /N
}

extern "C" void kernel_launch(void* const* d_in, const int* in_sizes, int n_in,
                              void* d_out, int out_size, void* d_ws, size_t ws_size,
                              hipStream_t stream) {
    const float* pred = (const float*)d_in[0];
    const float* tgt  = (const float*)d_in[1];
    const float* mask = (const float*)d_in[2];
    float* out = (float*)d_out;
    float* ws  = (float*)d_ws;
    const int N = 2;
    const int V = in_sizes[0] / N;          // 128^3

    dvh_init<<<5, 256, 0, stream>>>(ws);    // zero 1056 accumulator floats
    dim3 grid(1024, N);
    dvh_main<<<grid, 256, 0, stream>>>(pred, tgt, mask, ws, V);
    dvh_final<<<1, 512, 0, stream>>>(ws, out);
}